// Classifier_24008867184690
// MI455X (gfx1250) — compile-verified
//
#include <hip/hip_runtime.h>

// ---------------------------------------------------------------------------
// GCN 2-layer + graph mean-pool + linear classifier for MI455X (gfx1250).
// Layer-2 dense transform (100000x256 @ 256x128) uses V_WMMA_F32_16X16X4_F32
// (exact fp32 matrix pipe, wave32). Edge aggregations use L2-resident
// global_atomic_add_f32 scatter (h1/msum2 each 102MB < 192MB L2).
// ---------------------------------------------------------------------------

#define NN      100000
#define NE      1600000
#define NG      128
#define HD      256     // hidden 1
#define H2D     128     // hidden 2
#define CD      2

typedef float v2f __attribute__((ext_vector_type(2)));
typedef float v8f __attribute__((ext_vector_type(8)));

// ---- workspace layout (floats) --------------------------------------------
#define OFF_DEG   ((size_t)0)
#define OFF_A1    ((size_t)NN)                       // msum1, then a1 in place
#define OFF_MSUM2 ((size_t)(2 * NN))                 // msum2, then h_agg2 in place
#define OFF_GSUM  (OFF_MSUM2 + (size_t)NN * HD)
#define OFF_CNT   (OFF_GSUM + (size_t)NG * H2D)
#define OFF_H1    (OFF_CNT + (size_t)NG)             // h1, later reused as h2
#define ZERO_F    OFF_H1                             // floats to zero each launch

// ---------------------------------------------------------------------------
__global__ void zero_kernel(float4* __restrict__ p, int n4) {
    int i = blockIdx.x * blockDim.x + threadIdx.x;
    if (i < n4) p[i] = make_float4(0.f, 0.f, 0.f, 0.f);
}

// deg[dst[e]] += 1
__global__ void deg_kernel(const int* __restrict__ dst, float* __restrict__ deg) {
    int e = blockIdx.x * blockDim.x + threadIdx.x;
    if (e < NE) atomicAdd(&deg[dst[e]], 1.0f);
}

// msum1[dst[e]] += deg[src[e]]   (h0 == deg)
__global__ void agg1_kernel(const int* __restrict__ src, const int* __restrict__ dst,
                            const float* __restrict__ deg, float* __restrict__ msum1) {
    int e = blockIdx.x * blockDim.x + threadIdx.x;
    if (e < NE) atomicAdd(&msum1[dst[e]], deg[src[e]]);
}

// a1[n] = deg>0 ? msum1/deg : deg   (in place over msum1)
__global__ void a1_kernel(const float* __restrict__ deg, float* __restrict__ a1) {
    int n = blockIdx.x * blockDim.x + threadIdx.x;
    if (n < NN) {
        float d = deg[n];
        float m = a1[n];
        a1[n] = (d > 0.f) ? (m / d) : d;
    }
}

// h1[n][c] = relu(a1[n] * W1[c] + b1[c])   (rank-1 outer product, float4)
__global__ void h1_kernel(const float* __restrict__ a1,
                          const float4* __restrict__ W1, const float4* __restrict__ b1,
                          float4* __restrict__ h1) {
    int t = blockIdx.x * blockDim.x + threadIdx.x;   // NN * 64 threads
    int n = t >> 6, c4 = t & 63;
    float a = a1[n];
    float4 w = W1[c4], b = b1[c4];
    float4 r;
    r.x = fmaxf(a * w.x + b.x, 0.f);
    r.y = fmaxf(a * w.y + b.y, 0.f);
    r.z = fmaxf(a * w.z + b.z, 0.f);
    r.w = fmaxf(a * w.w + b.w, 0.f);
    h1[(size_t)n * 64 + c4] = r;
}

// msum2[dst[e]][:] += h1[src[e]][:]   (64 consecutive threads = one edge, float4)
__global__ void msum2_kernel(const int* __restrict__ src, const int* __restrict__ dst,
                             const float4* __restrict__ h1, float* __restrict__ msum2) {
    long long t = (long long)blockIdx.x * blockDim.x + threadIdx.x;  // NE * 64
    int e = (int)(t >> 6), c4 = (int)(t & 63);
    int s = src[e], d = dst[e];
    float4 v = h1[(size_t)s * 64 + c4];
    float* p = &msum2[(size_t)d * HD + c4 * 4];
    atomicAdd(p + 0, v.x);
    atomicAdd(p + 1, v.y);
    atomicAdd(p + 2, v.z);
    atomicAdd(p + 3, v.w);
}

// h_agg2 = deg>0 ? msum2/deg : h1   (in place over msum2)
__global__ void hagg2_kernel(const float* __restrict__ deg, const float4* __restrict__ h1,
                             float4* __restrict__ msum2) {
    int t = blockIdx.x * blockDim.x + threadIdx.x;   // NN * 64
    int n = t >> 6, c4 = t & 63;
    float d = deg[n];
    size_t i = (size_t)n * 64 + c4;
    float4 m = msum2[i];
    float4 h = h1[i];
    float4 r;
    if (d > 0.f) {
        float inv = 1.f / d;
        r.x = m.x * inv; r.y = m.y * inv; r.z = m.z * inv; r.w = m.w * inv;
    } else {
        r = h;
    }
    msum2[i] = r;
}

// h2 = relu(h_agg2 @ W2 + b2) via V_WMMA_F32_16X16X4_F32.
// Block = 256 threads (8 waves). Block handles 16 rows; wave w -> cols [16w,16w+16).
// A tile (16x256) staged in LDS with +4 float row padding (stride 260).
__global__ void __launch_bounds__(256)
gemm_wmma_kernel(const float* __restrict__ A,     // h_agg2, [NN][256]
                 const float* __restrict__ W2,    // [256][128]
                 const float* __restrict__ b2,    // [128]
                 float* __restrict__ h2) {        // [NN][128]
    __shared__ __align__(16) float As[16 * 260];

    const int tid  = threadIdx.x;
    const int row0 = blockIdx.x * 16;

    // cooperative A-tile load: 1024 float4s, fully coalesced
    const float4* A4 = (const float4*)(A + (size_t)row0 * HD);
#pragma unroll
    for (int j = 0; j < 4; ++j) {
        int idx4 = tid + j * 256;            // 0..1023
        int r = idx4 >> 6, c4 = idx4 & 63;   // 64 float4 per row
        *(float4*)&As[r * 260 + c4 * 4] = A4[idx4];
    }
    __syncthreads();

    const int lane = tid & 31;
    const int wave = tid >> 5;       // 0..7 -> col tile
    const int half = lane >> 4;      // 0: K+0/K+1, 1: K+2/K+3
    const int m    = lane & 15;
    const int col0 = wave * 16;

    v8f acc = {0.f, 0.f, 0.f, 0.f, 0.f, 0.f, 0.f, 0.f};

#pragma unroll 4
    for (int k0 = 0; k0 < HD; k0 += 4) {
        // A 16x4: lane<16 -> (M=m, K=k0..k0+1); lane>=16 -> (M=m, K=k0+2..k0+3)
        v2f a = *(const v2f*)&As[m * 260 + k0 + 2 * half];
        // B 4x16: vgpr0 = row K=k0+2*half, vgpr1 = row K=k0+2*half+1, N = m
        int kb = k0 + 2 * half;
        v2f b;
        b.x = W2[(size_t)kb * H2D + col0 + m];
        b.y = W2[(size_t)(kb + 1) * H2D + col0 + m];
        acc = __builtin_amdgcn_wmma_f32_16x16x4_f32(
            /*neg_a=*/false, a, /*neg_b=*/false, b,
            /*c_mod=*/(short)0, acc, /*reuse_a=*/false, /*reuse_b=*/false);
    }

    const float bias = b2[col0 + m];
    // D: vgpr v holds (M = v + 8*half, N = m)
#pragma unroll
    for (int v = 0; v < 8; ++v) {
        int row = row0 + v + half * 8;
        float x = acc[v] + bias;
        h2[(size_t)row * H2D + col0 + m] = fmaxf(x, 0.f);
    }
}

// gsum[g] += h2[n]; cnt[g] += 1
__global__ void pool_kernel(const int* __restrict__ gids, const float4* __restrict__ h2,
                            float* __restrict__ gsum, float* __restrict__ cnt) {
    int t = blockIdx.x * blockDim.x + threadIdx.x;   // NN * 32
    int n = t >> 5, c4 = t & 31;
    int g = gids[n];
    float4 v = h2[(size_t)n * 32 + c4];
    float* p = &gsum[(size_t)g * H2D + c4 * 4];
    atomicAdd(p + 0, v.x);
    atomicAdd(p + 1, v.y);
    atomicAdd(p + 2, v.z);
    atomicAdd(p + 3, v.w);
    if (c4 == 0) atomicAdd(&cnt[g], 1.0f);
}

// out[g][c] = (gsum[g]/max(cnt,1)) . Wc[:,c] + bc[c]
__global__ void out_kernel(const float* __restrict__ gsum, const float* __restrict__ cnt,
                           const float* __restrict__ Wc, const float* __restrict__ bc,
                           float* __restrict__ out) {
    int t = threadIdx.x;                 // 256 = G*C
    if (t < NG * CD) {
        int g = t >> 1, c = t & 1;
        float inv = 1.f / fmaxf(cnt[g], 1.f);
        float s = 0.f;
        for (int k = 0; k < H2D; ++k)
            s += gsum[(size_t)g * H2D + k] * Wc[k * CD + c];
        out[t] = s * inv + bc[c];
    }
}

// ---------------------------------------------------------------------------
extern "C" void kernel_launch(void* const* d_in, const int* in_sizes, int n_in,
                              void* d_out, int out_size, void* d_ws, size_t ws_size,
                              hipStream_t stream) {
    (void)in_sizes; (void)n_in; (void)out_size; (void)ws_size;
    const int*   src  = (const int*)d_in[0];
    const int*   dst  = (const int*)d_in[1];
    const int*   gids = (const int*)d_in[2];
    const float* W1   = (const float*)d_in[3];
    const float* b1   = (const float*)d_in[4];
    const float* W2   = (const float*)d_in[5];
    const float* b2   = (const float*)d_in[6];
    const float* Wc   = (const float*)d_in[7];
    const float* bc   = (const float*)d_in[8];
    float* out = (float*)d_out;
    float* ws  = (float*)d_ws;

    float* deg   = ws + OFF_DEG;
    float* a1    = ws + OFF_A1;     // msum1 then a1
    float* msum2 = ws + OFF_MSUM2;  // msum2 then h_agg2
    float* gsum  = ws + OFF_GSUM;
    float* cnt   = ws + OFF_CNT;
    float* h1    = ws + OFF_H1;     // h1 then h2

    // 1) zero accumulators (deg, msum1, msum2, gsum, cnt)
    {
        int n4 = (int)(ZERO_F / 4);
        zero_kernel<<<(n4 + 255) / 256, 256, 0, stream>>>((float4*)ws, n4);
    }
    // 2) degree
    deg_kernel<<<NE / 256, 256, 0, stream>>>(dst, deg);
    // 3) layer-1 scalar aggregation
    agg1_kernel<<<NE / 256, 256, 0, stream>>>(src, dst, deg, a1);
    // 4) layer-1 mean / where
    a1_kernel<<<(NN + 255) / 256, 256, 0, stream>>>(deg, a1);
    // 5) h1 = relu(a1 * W1 + b1)
    h1_kernel<<<(NN * 64) / 256, 256, 0, stream>>>(
        a1, (const float4*)W1, (const float4*)b1, (float4*)h1);
    // 6) layer-2 aggregation (scatter-add, 64 threads per edge)
    msum2_kernel<<<(NE * 64) / 256, 256, 0, stream>>>(src, dst, (const float4*)h1, msum2);
    // 7) h_agg2 = where(deg>0, msum2/deg, h1)  (in place)
    hagg2_kernel<<<(NN * 64) / 256, 256, 0, stream>>>(deg, (const float4*)h1, (float4*)msum2);
    // 8) h2 = relu(h_agg2 @ W2 + b2) via WMMA f32 16x16x4 (h2 reuses h1 buffer)
    gemm_wmma_kernel<<<NN / 16, 256, 0, stream>>>(msum2, W2, b2, h1);
    // 9) graph mean-pool accumulate
    pool_kernel<<<(NN * 32) / 256, 256, 0, stream>>>(gids, (const float4*)h1, gsum, cnt);
    // 10) classifier
    out_kernel<<<1, 256, 0, stream>>>(gsum, cnt, Wc, bc, out);
}